// EdgeNet_Ori_75900662055231
// MI455X (gfx1250) — compile-verified
//
#include <hip/hip_runtime.h>

typedef __attribute__((ext_vector_type(2))) float v2f;
typedef __attribute__((ext_vector_type(8))) float v8f;

static __device__ __forceinline__ v8f wmma4(v2f a, v2f b, v8f c) {
  // D = A(16x4,f32) * B(4x16,f32) + C(16x16,f32)   -> v_wmma_f32_16x16x4_f32
  return __builtin_amdgcn_wmma_f32_16x16x4_f32(false, a, false, b, (short)0, c, false, false);
}

constexpr int NN = 50000;   // nodes
constexpr int EE = 800000;  // edges
constexpr int DD = 50;      // input feature dim
constexpr int HH = 50;      // hidden dim
constexpr int OO = 100;     // output dim

// ------------------------------------------------------------------ counting
__global__ void k_count(const int* __restrict__ col, int* __restrict__ cnt) {
  int i = blockIdx.x * blockDim.x + threadIdx.x;
  if (i < EE) atomicAdd(&cnt[col[i]], 1);
}

// ------------------------------------------------- GEMM 1: xW1a = x @ W1[:50]
__global__ void k_gemm_xW1a(const float* __restrict__ x, const float* __restrict__ W1,
                            float* __restrict__ xW1a) {
  constexpr int KP = 52, NP = 64;
  __shared__ float Wl[KP * NP];
  for (int i = threadIdx.x; i < KP * NP; i += blockDim.x) {
    int kk = i / NP, nn = i % NP;
    Wl[i] = (kk < DD && nn < HH) ? W1[kk * HH + nn] : 0.f;
  }
  __syncthreads();
  const int lane = threadIdx.x & 31, wv = threadIdx.x >> 5;
  const int lm = lane & 15, lh = lane >> 4;
  const int m0 = (blockIdx.x * 8 + wv) * 16;
  const int arow = min(m0 + lm, NN - 1);
  const float* __restrict__ ap = x + (size_t)arow * DD;
  v8f c[4] = {};
  for (int k = 0; k < DD; k += 4) {
    int kk = k + 2 * lh;
    v2f a = {0.f, 0.f};
    if (kk < DD) { a[0] = ap[kk]; a[1] = ap[kk + 1]; }
#pragma unroll
    for (int t = 0; t < 4; ++t) {
      v2f b; int n = t * 16 + lm;
      b[0] = Wl[kk * NP + n]; b[1] = Wl[(kk + 1) * NP + n];
      c[t] = wmma4(a, b, c[t]);
    }
  }
#pragma unroll
  for (int t = 0; t < 4; ++t) {
    int n = t * 16 + lm;
    if (n < HH) {
#pragma unroll
      for (int i = 0; i < 8; ++i) {
        int m = m0 + i + 8 * lh;
        if (m < NN) xW1a[(size_t)m * HH + n] = c[t][i];
      }
    }
  }
}

// ------------- GEMM 2: h1head[e] = edge_attr[e] @ W1[50:] + xW1a[row[e]], e<N
__global__ void k_gemm_h1head(const float* __restrict__ ea, const float* __restrict__ W1,
                              const int* __restrict__ row, const float* __restrict__ xW1a,
                              float* __restrict__ h1head) {
  constexpr int KP = 52, NP = 64;
  __shared__ float Wl[KP * NP];
  const float* __restrict__ W1b = W1 + DD * HH;
  for (int i = threadIdx.x; i < KP * NP; i += blockDim.x) {
    int kk = i / NP, nn = i % NP;
    Wl[i] = (kk < DD && nn < HH) ? W1b[kk * HH + nn] : 0.f;
  }
  __syncthreads();
  const int lane = threadIdx.x & 31, wv = threadIdx.x >> 5;
  const int lm = lane & 15, lh = lane >> 4;
  const int m0 = (blockIdx.x * 8 + wv) * 16;
  const int arow = min(m0 + lm, NN - 1);
  const float* __restrict__ ap = ea + (size_t)arow * DD;
  v8f c[4] = {};
  for (int k = 0; k < DD; k += 4) {
    int kk = k + 2 * lh;
    v2f a = {0.f, 0.f};
    if (kk < DD) { a[0] = ap[kk]; a[1] = ap[kk + 1]; }
#pragma unroll
    for (int t = 0; t < 4; ++t) {
      v2f b; int n = t * 16 + lm;
      b[0] = Wl[kk * NP + n]; b[1] = Wl[(kk + 1) * NP + n];
      c[t] = wmma4(a, b, c[t]);
    }
  }
#pragma unroll
  for (int t = 0; t < 4; ++t) {
    int n = t * 16 + lm;
    if (n < HH) {
#pragma unroll
      for (int i = 0; i < 8; ++i) {
        int m = m0 + i + 8 * lh;
        if (m < NN) {
          int r = row[m];
          h1head[(size_t)m * HH + n] = c[t][i] + xW1a[(size_t)r * HH + n];
        }
      }
    }
  }
}

// -------------------- agg[col[e]] += h1head[row[e]] * rsqrt(deg1(row[e]))
__global__ void k_scatter_agg(const int* __restrict__ row, const int* __restrict__ col,
                              const int* __restrict__ cnt, const float* __restrict__ h1head,
                              float* __restrict__ agg) {
  int i = blockIdx.x * blockDim.x + threadIdx.x;
  if (i >= EE * HH) return;
  int e = i / HH, k = i - e * HH;
  int r = row[e], c = col[e];
  float dr = rsqrtf((float)(cnt[r] + 1));
  atomicAdd(&agg[(size_t)c * HH + k], h1head[(size_t)r * HH + k] * dr);
}

// --- GEMM 3 over all E edges: recompute h1, finish GCN1 + relu, scatter_mean sums
__global__ void k_gemm_r1(const float* __restrict__ ea, const float* __restrict__ W1,
                          const int* __restrict__ row, const int* __restrict__ col,
                          const int* __restrict__ cnt, const float* __restrict__ xW1a,
                          const float* __restrict__ agg, const float* __restrict__ b1,
                          float* __restrict__ sums) {
  constexpr int KP = 52, NP = 64;
  __shared__ float Wl[KP * NP];
  const float* __restrict__ W1b = W1 + DD * HH;
  for (int i = threadIdx.x; i < KP * NP; i += blockDim.x) {
    int kk = i / NP, nn = i % NP;
    Wl[i] = (kk < DD && nn < HH) ? W1b[kk * HH + nn] : 0.f;
  }
  __syncthreads();
  const int lane = threadIdx.x & 31, wv = threadIdx.x >> 5;
  const int lm = lane & 15, lh = lane >> 4;
  const int m0 = (blockIdx.x * 8 + wv) * 16;
  const int arow = min(m0 + lm, EE - 1);
  const float* __restrict__ ap = ea + (size_t)arow * DD;
  v8f c[4] = {};
  for (int k = 0; k < DD; k += 4) {
    int kk = k + 2 * lh;
    v2f a = {0.f, 0.f};
    if (kk < DD) { a[0] = ap[kk]; a[1] = ap[kk + 1]; }
#pragma unroll
    for (int t = 0; t < 4; ++t) {
      v2f b; int n = t * 16 + lm;
      b[0] = Wl[kk * NP + n]; b[1] = Wl[(kk + 1) * NP + n];
      c[t] = wmma4(a, b, c[t]);
    }
  }
#pragma unroll
  for (int t = 0; t < 4; ++t) {
    int n = t * 16 + lm;
    if (n < HH) {
      float bn = b1[n];
#pragma unroll
      for (int i = 0; i < 8; ++i) {
        int e = m0 + i + 8 * lh;
        if (e < EE) {
          int r = row[e];
          float h = c[t][i] + xW1a[(size_t)r * HH + n];
          float val;
          if (e < NN) {
            float d = (float)(cnt[e] + 1);      // deg1 of E-graph "node" e
            val = bn + h * (1.f / d) + rsqrtf(d) * agg[(size_t)e * HH + n];
          } else {
            val = bn + h;                        // deg1 == 1 (self loop only)
          }
          val = fmaxf(val, 0.f);
          atomicAdd(&sums[(size_t)col[e] * HH + n], val);
        }
      }
    }
  }
}

// -------- GEMM 4: h2 = concat(x, sums*invcnt) @ W2   (K = 100, N-tiles = 7)
__global__ void k_gemm_h2(const float* __restrict__ x, const float* __restrict__ sums,
                          const int* __restrict__ cnt, const float* __restrict__ W2,
                          float* __restrict__ h2) {
  constexpr int KP = 100, NP = 112;
  __shared__ float Wl[KP * NP];
  for (int i = threadIdx.x; i < KP * NP; i += blockDim.x) {
    int kk = i / NP, nn = i % NP;
    Wl[i] = (nn < OO) ? W2[kk * OO + nn] : 0.f;
  }
  __syncthreads();
  const int lane = threadIdx.x & 31, wv = threadIdx.x >> 5;
  const int lm = lane & 15, lh = lane >> 4;
  const int m0 = (blockIdx.x * 8 + wv) * 16;
  const int arow = min(m0 + lm, NN - 1);
  const float* __restrict__ xr = x + (size_t)arow * DD;
  const float* __restrict__ mr = sums + (size_t)arow * HH;
  const float ic = 1.f / fmaxf((float)cnt[arow], 1.f);   // scatter_mean divisor
  v8f c[7] = {};
  for (int k = 0; k < KP; k += 4) {
    int kk = k + 2 * lh;
    v2f a;
    if (kk < DD) { a[0] = xr[kk];           a[1] = xr[kk + 1]; }
    else         { a[0] = mr[kk - DD] * ic; a[1] = mr[kk - DD + 1] * ic; }
#pragma unroll
    for (int t = 0; t < 7; ++t) {
      v2f b; int n = t * 16 + lm;
      b[0] = Wl[kk * NP + n]; b[1] = Wl[(kk + 1) * NP + n];
      c[t] = wmma4(a, b, c[t]);
    }
  }
#pragma unroll
  for (int t = 0; t < 7; ++t) {
    int n = t * 16 + lm;
    if (n < OO) {
#pragma unroll
      for (int i = 0; i < 8; ++i) {
        int m = m0 + i + 8 * lh;
        if (m < NN) h2[(size_t)m * OO + n] = c[t][i];
      }
    }
  }
}

// -------------------- agg2[col[e]] += h2[row[e]] * rsqrt(deg2(row[e]))
__global__ void k_scatter_agg2(const int* __restrict__ row, const int* __restrict__ col,
                               const int* __restrict__ cnt, const float* __restrict__ h2,
                               float* __restrict__ agg2) {
  int i = blockIdx.x * blockDim.x + threadIdx.x;
  if (i >= EE * OO) return;
  int e = i / OO, k = i - e * OO;
  int r = row[e], c = col[e];
  float dr = rsqrtf((float)(cnt[r] + 1));
  atomicAdd(&agg2[(size_t)c * OO + k], h2[(size_t)r * OO + k] * dr);
}

// ----------- finish GCN2 + relu + column mean over N rows -> out[100]
__global__ void k_final(const float* __restrict__ h2, const float* __restrict__ agg2,
                        const int* __restrict__ cnt, const float* __restrict__ b2,
                        float* __restrict__ out) {
  const int n = threadIdx.x;
  if (n >= OO) return;
  int j0 = blockIdx.x * 512;
  int j1 = min(j0 + 512, NN);
  float bn = b2[n];
  float acc = 0.f;
  for (int j = j0; j < j1; ++j) {
    float d = (float)(cnt[j] + 1);
    float v = bn + h2[(size_t)j * OO + n] * (1.f / d) + rsqrtf(d) * agg2[(size_t)j * OO + n];
    acc += fmaxf(v, 0.f);
  }
  atomicAdd(&out[n], acc * (1.f / (float)NN));
}

// --------------------------------------------------------------------- launch
static constexpr size_t algn(size_t x) { return (x + 255) & ~(size_t)255; }

extern "C" void kernel_launch(void* const* d_in, const int* in_sizes, int n_in,
                              void* d_out, int out_size, void* d_ws, size_t ws_size,
                              hipStream_t stream) {
  (void)in_sizes; (void)n_in; (void)out_size;
  const float* x   = (const float*)d_in[0];
  const float* ea  = (const float*)d_in[1];
  const float* W1  = (const float*)d_in[2];
  const float* b1  = (const float*)d_in[3];
  const float* W2  = (const float*)d_in[4];
  const float* b2  = (const float*)d_in[5];
  const int*   ei  = (const int*)d_in[6];
  const int*   row = ei;
  const int*   col = ei + EE;

  char* w = (char*)d_ws;
  constexpr size_t OFF_CNT  = 0;
  constexpr size_t OFF_XW1A = algn(OFF_CNT  + (size_t)NN * sizeof(int));
  constexpr size_t OFF_H1   = algn(OFF_XW1A + (size_t)NN * HH * sizeof(float));
  constexpr size_t OFF_AGG  = algn(OFF_H1   + (size_t)NN * HH * sizeof(float));
  constexpr size_t OFF_SUMS = algn(OFF_AGG  + (size_t)NN * HH * sizeof(float));
  constexpr size_t OFF_H2   = algn(OFF_SUMS + (size_t)NN * HH * sizeof(float));
  constexpr size_t OFF_AGG2 = algn(OFF_H2   + (size_t)NN * OO * sizeof(float));
  (void)ws_size;  // needs ~80.2 MB

  int*   cnt   = (int*)  (w + OFF_CNT);
  float* xW1a  = (float*)(w + OFF_XW1A);
  float* h1hd  = (float*)(w + OFF_H1);
  float* agg   = (float*)(w + OFF_AGG);
  float* sums  = (float*)(w + OFF_SUMS);
  float* h2    = (float*)(w + OFF_H2);
  float* agg2  = (float*)(w + OFF_AGG2);
  float* out   = (float*)d_out;

  hipMemsetAsync(cnt,  0, (size_t)NN * sizeof(int),        stream);
  hipMemsetAsync(agg,  0, (size_t)NN * HH * sizeof(float), stream);
  hipMemsetAsync(sums, 0, (size_t)NN * HH * sizeof(float), stream);
  hipMemsetAsync(agg2, 0, (size_t)NN * OO * sizeof(float), stream);
  hipMemsetAsync(out,  0, (size_t)OO * sizeof(float),      stream);

  k_count      <<<(EE + 255) / 256, 256, 0, stream>>>(col, cnt);
  k_gemm_xW1a  <<<(NN + 127) / 128, 256, 0, stream>>>(x, W1, xW1a);
  k_gemm_h1head<<<(NN + 127) / 128, 256, 0, stream>>>(ea, W1, row, xW1a, h1hd);
  k_scatter_agg<<<(EE * HH + 255) / 256, 256, 0, stream>>>(row, col, cnt, h1hd, agg);
  k_gemm_r1    <<<(EE + 127) / 128, 256, 0, stream>>>(ea, W1, row, col, cnt, xW1a, agg, b1, sums);
  k_gemm_h2    <<<(NN + 127) / 128, 256, 0, stream>>>(x, sums, cnt, W2, h2);
  k_scatter_agg2<<<(EE * OO + 255) / 256, 256, 0, stream>>>(row, col, cnt, h2, agg2);
  k_final      <<<(NN + 511) / 512, 128, 0, stream>>>(h2, agg2, cnt, b2, out);
}